// FeatureHE_81355270521051
// MI455X (gfx1250) — compile-verified
//
#include <hip/hip_runtime.h>

typedef __attribute__((ext_vector_type(2))) float v2f;
typedef __attribute__((ext_vector_type(8))) float v8f;

#define NBINS      64
#define NPIX       16384      // 128*128 pixels per (b,c) channel
#define THREADS    256
#define VPT        16         // float4 per thread: 16*4*256 = 16384
#define ROWSTRIDE  65         // padded LDS row stride (bank-conflict free)

// LDS layout (floats)
#define OFF_HIST   0                    // 256*65 = 16640
#define OFF_FINAL  16640                // 64
#define OFF_CDF    16704                // 64
#define OFF_RED    16768                // 16 (8 min + 8 max)
#define OFF_SCAL   16784                // 4  (mn, rng, 1/rng)
#define SMEM_FLOATS 16788

__global__ __launch_bounds__(THREADS) void featurehe_kernel(
    const float* __restrict__ x,
    const float* __restrict__ p_log_sigma,
    const float* __restrict__ p_alpha,
    float* __restrict__ out)
{
    extern __shared__ float smem[];
    float* histp  = smem + OFF_HIST;
    float* finalh = smem + OFF_FINAL;
    float* cdf    = smem + OFF_CDF;
    float* red    = smem + OFF_RED;
    float* scal   = smem + OFF_SCAL;

    const int tid  = threadIdx.x;
    const int lane = tid & 31;
    const int wv   = tid >> 5;
    const int ch   = blockIdx.x;

    // ---- Load 64 pixels/thread into registers; track min/max -------------
    const float4* xv = (const float4*)(x + (size_t)ch * NPIX);
    float4 px[VPT];
    float vmin = 3.4e38f, vmax = -3.4e38f;
#pragma unroll
    for (int i = 0; i < VPT; ++i) {
        float4 v = xv[tid + i * THREADS];
        px[i] = v;
        vmin = fminf(vmin, fminf(fminf(v.x, v.y), fminf(v.z, v.w)));
        vmax = fmaxf(vmax, fmaxf(fmaxf(v.x, v.y), fmaxf(v.z, v.w)));
    }
    // wave32 butterfly reduce
#pragma unroll
    for (int off = 16; off > 0; off >>= 1) {
        vmin = fminf(vmin, __shfl_xor(vmin, off, 32));
        vmax = fmaxf(vmax, __shfl_xor(vmax, off, 32));
    }
    if (lane == 0) { red[wv] = vmin; red[8 + wv] = vmax; }
    __syncthreads();
    if (tid == 0) {
        float mn = red[0], mx = red[8];
        for (int i = 1; i < 8; ++i) { mn = fminf(mn, red[i]); mx = fmaxf(mx, red[8 + i]); }
        float rng = mx - mn + 1e-6f;
        scal[0] = mn; scal[1] = rng; scal[2] = 1.0f / rng;
    }
    // zero this thread's private histogram row + the final histogram
    float* rowp = histp + tid * ROWSTRIDE;
#pragma unroll
    for (int k = 0; k < ROWSTRIDE; ++k) rowp[k] = 0.0f;
    if (tid < NBINS) finalh[tid] = 0.0f;
    __syncthreads();

    const float mn = scal[0], rng = scal[1], invRng = scal[2];

    // sigma / Gaussian params (scalar inputs, computed redundantly per thread)
    const float ls    = p_log_sigma[0];
    const float sigma = log1pf(__expf(ls)) + 1e-6f;
    const float inv   = -0.5f / (sigma * sigma + 1e-12f);
    const float c2    = inv * 1.4426950408889634f;   // fold log2(e) for v_exp_f32
    // truncate where |inv| * d^2 > 14  (exp < 8.3e-7 of peak)
    int radius = (int)ceilf(63.0f * __builtin_sqrtf(14.0f / fabsf(inv)));
    radius = radius < 1 ? 1 : (radius > 63 ? 63 : radius);

    // ---- Soft histogram into private LDS row -----------------------------
#pragma unroll
    for (int i = 0; i < VPT; ++i) {
        float vv[4] = { px[i].x, px[i].y, px[i].z, px[i].w };
#pragma unroll
        for (int j = 0; j < 4; ++j) {
            float xn  = (vv[j] - mn) * invRng;     // in [0,1)
            float pos = xn * 63.0f;
            int ib = (int)pos;
            int k0 = ib - radius; if (k0 < 0)  k0 = 0;
            int k1 = ib + radius; if (k1 > 63) k1 = 63;
            for (int k = k0; k <= k1; ++k) {
                float d = xn - (float)k * (1.0f / 63.0f);
                rowp[k] += __builtin_amdgcn_exp2f(c2 * d * d);
            }
        }
    }
    __syncthreads();

    // ---- Column-sum the 256x64 partial-hist matrix with WMMA f32 16x16x4 -
    // Each wave reduces its 32 rows (8 chunks of K=4) per 16-bin tile:
    //   D += ones(16x4) x B(4x16)  -> every row of D holds the column sums.
    v2f onesA; onesA[0] = 1.0f; onesA[1] = 1.0f;
    const int rsel = (lane >> 4) << 1;      // B-matrix lane->row split: +0 / +2
#pragma unroll
    for (int t = 0; t < 4; ++t) {
        v8f acc = {0.f, 0.f, 0.f, 0.f, 0.f, 0.f, 0.f, 0.f};
#pragma unroll
        for (int cidx = 0; cidx < 8; ++cidx) {
            int r0 = wv * 32 + cidx * 4 + rsel;
            const float* bp = histp + r0 * ROWSTRIDE + t * 16 + (lane & 15);
            v2f bb; bb[0] = bp[0]; bb[1] = bp[ROWSTRIDE];
            acc = __builtin_amdgcn_wmma_f32_16x16x4_f32(
                /*neg_a=*/false, onesA, /*neg_b=*/false, bb,
                /*c_mod=*/(short)0, acc, /*reuse_a=*/false, /*reuse_b=*/false);
        }
        if (lane < 16) atomicAdd(&finalh[t * 16 + lane], acc[0]);
    }
    __syncthreads();

    // ---- Normalize + CDF (64 serial steps, one thread) -------------------
    if (tid == 0) {
        float total = 0.0f;
        for (int k = 0; k < NBINS; ++k) total += finalh[k];
        float s = 1.0f / (total + 1e-6f);
        float run = 0.0f;
        for (int k = 0; k < NBINS; ++k) { run += finalh[k] * s; cdf[k] = run; }
        float c0 = cdf[0];
        float is = 1.0f / (cdf[NBINS - 1] - c0 + 1e-6f);
        for (int k = 0; k < NBINS; ++k) cdf[k] = (cdf[k] - c0) * is;
    }
    __syncthreads();

    // ---- Interp into CDF + blend, straight from registers ----------------
    const float alpha = p_alpha[0];
    const float a  = 1.0f / (1.0f + __expf(-alpha));
    const float na = 1.0f - a;
    float4* ov = (float4*)(out + (size_t)ch * NPIX);

#define EQ1(val, res) {                                              \
        float xn = (val - mn) * invRng;                              \
        xn = xn < 0.0f ? 0.0f : (xn > 1.0f ? 1.0f : xn);             \
        float pos = xn * 63.0f;                                      \
        int idx = (int)pos; if (idx > 62) idx = 62;                  \
        float fr = pos - (float)idx;                                 \
        float lo = cdf[idx], hi = cdf[idx + 1];                      \
        float xeq = (lo + fr * (hi - lo)) * rng + mn;                \
        res = a * xeq + na * (val);                                  \
    }

#pragma unroll
    for (int i = 0; i < VPT; ++i) {
        float4 v = px[i], o;
        EQ1(v.x, o.x); EQ1(v.y, o.y); EQ1(v.z, o.z); EQ1(v.w, o.w);
        ov[tid + i * THREADS] = o;
    }
#undef EQ1
}

extern "C" void kernel_launch(void* const* d_in, const int* in_sizes, int n_in,
                              void* d_out, int out_size, void* d_ws, size_t ws_size,
                              hipStream_t stream) {
    (void)n_in; (void)d_ws; (void)ws_size; (void)out_size;
    const float* x         = (const float*)d_in[0];
    const float* log_sigma = (const float*)d_in[1];
    const float* alpha     = (const float*)d_in[2];
    float* out             = (float*)d_out;

    const int nchan = in_sizes[0] / NPIX;            // B*C = 2048
    const size_t shbytes = (size_t)SMEM_FLOATS * sizeof(float);  // ~67 KB of the 320 KB LDS
    featurehe_kernel<<<nchan, THREADS, shbytes, stream>>>(x, log_sigma, alpha, out);
}